// SIRD_layer_22565758173431
// MI455X (gfx1250) — compile-verified
//
#include <hip/hip_runtime.h>

// ---------------- Tsit5 tableau (fp32, matching the reference) ----------------
#define C_ATOL 1e-8f
#define C_RTOL 1e-4f
#define SAFETY 0.9f
#define MIN_FAC 0.2f
#define MAX_FAC 10.0f
#define PRED_LEN 32
#define MAX_STEPS 12

#define A21 0.161f
#define A31 (-0.008480655492356989f)
#define A32 0.335480655492357f
#define A41 2.8971530571054935f
#define A42 (-6.359448489975075f)
#define A43 4.3622954328695815f
#define A51 5.325864828439257f
#define A52 (-11.748883564062828f)
#define A53 7.4955393428898365f
#define A54 (-0.09249506636175525f)
#define A61 5.86145544294642f
#define A62 (-12.92096931784711f)
#define A63 8.159367898576159f
#define A64 (-0.071584973281401f)
#define A65 (-0.028269050394068383f)
#define Bc1 0.09646076681806523f
#define Bc2 0.01f
#define Bc3 0.4798896504144996f
#define Bc4 1.379008574103742f
#define Bc5 (-3.290069515436081f)
#define Bc6 2.324710524099774f
#define E1 (-0.001780011052225777f)
#define E2 (-0.0008164344596567469f)
#define E3 0.007880878010261995f
#define E4 (-0.1447110071732629f)
#define E5 0.5823571654525552f
#define E6 (-0.45808210592918697f)
#define E7 0.015151515151515152f

// jnp.linspace(0, 32, 32) in f32: start + i*step, endpoint forced to stop.
__device__ __forceinline__ float teval(int i) {
    return (i == PRED_LEN - 1) ? 32.0f : (float)i * (32.0f / 31.0f);
}

__global__ __launch_bounds__(256) void sird_tsit5_kernel(
    const float* __restrict__ x,    // [B,3] beta,gamma,mu
    const float* __restrict__ gp,   // [B,1]
    const float* __restrict__ pop,  // [B]
    float4* __restrict__ out,       // [B, PRED_LEN] as float4 (S,I,R,D)
    int B)
{
    const int b = blockIdx.x * blockDim.x + threadIdx.x;
    if (b >= B) return;

    // gfx1250 prefetch path (global_prefetch_b8) for the parameter cachelines.
    __builtin_prefetch(&x[3 * b], 0, 3);
    __builtin_prefetch(&pop[b], 0, 3);

    const float beta  = x[3 * b + 0];
    const float gamma = x[3 * b + 1];
    const float mu    = x[3 * b + 2];
    const float N     = pop[b];
    const float bN    = beta / N;     // beta/N folded once (precise div, cold)
    const float gmu   = gamma + mu;

    float S = gp[b] - 100.0f;
    float I = 100.0f;
    float R = 0.0f;
    float D = 0.0f;
    float dt = 0.05f;

    const size_t obase = (size_t)b * PRED_LEN;
    out[obase] = make_float4(S, I, R, D);

    // rhs depends only on (S,I): kS = -inf, kI = inf - (g+mu)*I, with inf = (beta/N)*S*I.
    // kR = gamma*I_stage, kD = mu*I_stage are folded analytically below.
#define RHS(Sv, Iv, ks, ki)                    \
    {                                          \
        float _inf = bN * (Sv) * (Iv);         \
        (ks) = -_inf;                          \
        (ki) = fmaf(-gmu, (Iv), _inf);         \
    }

#pragma unroll 1
    for (int seg = 0; seg < PRED_LEN - 1; ++seg) {
        float t = teval(seg);
        const float t1 = teval(seg + 1);

#pragma unroll 1
        for (int it = 0; it < MAX_STEPS; ++it) {
            const float h = fmaxf(fminf(dt, t1 - t), 1e-9f);

            float k1s, k1i, k2s, k2i, k3s, k3i, k4s, k4i, k5s, k5i, k6s, k6i, k7s, k7i;
            const float I1 = I;
            RHS(S, I, k1s, k1i);

            float cS = A21 * k1s;
            float cI = A21 * k1i;
            const float S2 = fmaf(h, cS, S);
            const float I2 = fmaf(h, cI, I);
            RHS(S2, I2, k2s, k2i);

            cS = fmaf(A32, k2s, A31 * k1s);
            cI = fmaf(A32, k2i, A31 * k1i);
            const float S3 = fmaf(h, cS, S);
            const float I3 = fmaf(h, cI, I);
            RHS(S3, I3, k3s, k3i);

            cS = fmaf(A43, k3s, fmaf(A42, k2s, A41 * k1s));
            cI = fmaf(A43, k3i, fmaf(A42, k2i, A41 * k1i));
            const float S4 = fmaf(h, cS, S);
            const float I4 = fmaf(h, cI, I);
            RHS(S4, I4, k4s, k4i);

            cS = fmaf(A54, k4s, fmaf(A53, k3s, fmaf(A52, k2s, A51 * k1s)));
            cI = fmaf(A54, k4i, fmaf(A53, k3i, fmaf(A52, k2i, A51 * k1i)));
            const float S5 = fmaf(h, cS, S);
            const float I5 = fmaf(h, cI, I);
            RHS(S5, I5, k5s, k5i);

            cS = fmaf(A65, k5s, fmaf(A64, k4s, fmaf(A63, k3s, fmaf(A62, k2s, A61 * k1s))));
            cI = fmaf(A65, k5i, fmaf(A64, k4i, fmaf(A63, k3i, fmaf(A62, k2i, A61 * k1i))));
            const float S6 = fmaf(h, cS, S);
            const float I6 = fmaf(h, cI, I);
            RHS(S6, I6, k6s, k6i);

            // 5th-order solution
            const float bS  = fmaf(Bc6, k6s, fmaf(Bc5, k5s, fmaf(Bc4, k4s, fmaf(Bc3, k3s, fmaf(Bc2, k2s, Bc1 * k1s)))));
            const float bI  = fmaf(Bc6, k6i, fmaf(Bc5, k5i, fmaf(Bc4, k4i, fmaf(Bc3, k3i, fmaf(Bc2, k2i, Bc1 * k1i)))));
            const float bII = fmaf(Bc6, I6,  fmaf(Bc5, I5,  fmaf(Bc4, I4,  fmaf(Bc3, I3,  fmaf(Bc2, I2,  Bc1 * I1)))));
            const float hg = h * gamma;
            const float hm = h * mu;
            const float ynS = fmaf(h, bS, S);
            const float ynI = fmaf(h, bI, I);
            const float ynR = fmaf(hg, bII, R);
            const float ynD = fmaf(hm, bII, D);

            RHS(ynS, ynI, k7s, k7i);
            const float I7 = ynI;

            // embedded error estimate
            const float eS  = fmaf(E7, k7s, fmaf(E6, k6s, fmaf(E5, k5s, fmaf(E4, k4s, fmaf(E3, k3s, fmaf(E2, k2s, E1 * k1s))))));
            const float eI  = fmaf(E7, k7i, fmaf(E6, k6i, fmaf(E5, k5i, fmaf(E4, k4i, fmaf(E3, k3i, fmaf(E2, k2i, E1 * k1i))))));
            const float eII = fmaf(E7, I7,  fmaf(E6, I6,  fmaf(E5, I5,  fmaf(E4, I4,  fmaf(E3, I3,  fmaf(E2, I2,  E1 * I1))))));
            const float errS = h * eS;
            const float errI = h * eI;
            const float errR = hg * eII;
            const float errD = hm * eII;

            // tol > ATOL > 0 always -> v_rcp_f32 (TRANS, co-executes) instead of
            // the ~10-op correctly-rounded divide sequence; feeds only the
            // threshold test + step controller, 1-ulp error is immaterial.
            const float tS = fmaf(C_RTOL, fmaxf(fabsf(S), fabsf(ynS)), C_ATOL);
            const float tI = fmaf(C_RTOL, fmaxf(fabsf(I), fabsf(ynI)), C_ATOL);
            const float tR = fmaf(C_RTOL, fmaxf(fabsf(R), fabsf(ynR)), C_ATOL);
            const float tD = fmaf(C_RTOL, fmaxf(fabsf(D), fabsf(ynD)), C_ATOL);

            float q = errS * __builtin_amdgcn_rcpf(tS);
            float ss = q * q;
            q = errI * __builtin_amdgcn_rcpf(tI); ss = fmaf(q, q, ss);
            q = errR * __builtin_amdgcn_rcpf(tR); ss = fmaf(q, q, ss);
            q = errD * __builtin_amdgcn_rcpf(tD); ss = fmaf(q, q, ss);

            // enorm = sqrt(ss/4);  enorm <= 1  <=>  ss <= 4   (sqrt folded away)
            // fac = 0.9*enorm^-0.2 = 0.9*(ss/4)^-0.1 ; enorm==0 branch -> 90 -> clamps to 10
            float fac;
            if (ss > 0.0f) {
                fac = SAFETY * exp2f(-0.1f * log2f(0.25f * ss));
                fac = fminf(fmaxf(fac, MIN_FAC), MAX_FAC);
            } else {
                fac = MAX_FAC;
            }

            if (ss <= 4.0f) {              // accept (done is false here by construction)
                t += h;
                S = ynS; I = ynI; R = ynR; D = ynD;
            }
            dt = h * fac;                  // dt updated before 'done' check, as in the scan
            if (t >= t1 - 1e-6f) break;    // frozen thereafter in the reference => exact early exit
        }

        out[obase + seg + 1] = make_float4(S, I, R, D);
    }
#undef RHS
}

extern "C" void kernel_launch(void* const* d_in, const int* in_sizes, int n_in,
                              void* d_out, int out_size, void* d_ws, size_t ws_size,
                              hipStream_t stream) {
    const float* x   = (const float*)d_in[0];  // [B,3]
    const float* gp  = (const float*)d_in[1];  // [B,1]
    const float* pop = (const float*)d_in[2];  // [B]
    float4* out = (float4*)d_out;

    const int B = in_sizes[0] / 3;
    dim3 block(256);
    dim3 grid((B + 255) / 256);
    hipLaunchKernelGGL(sird_tsit5_kernel, grid, block, 0, stream, x, gp, pop, out, B);
}